// HypAgg_43877385896091
// MI455X (gfx1250) — compile-verified
//
#include <hip/hip_runtime.h>
#include <math.h>

// HypAgg fused kernel for MI455X (gfx1250, wave32).
//
// Factorization: logmap(x_i,x_j) = p_ij*x_i + q_ij*x_j with scalars derived
// from the Gram matrix G = X X^T. Two fp32 WMMA GEMMs (16x16x4) + an N^2
// transcendental pass, fully fused per 16-row block. Only HBM traffic is one
// pass over adj (9.4 MB) -> ~0.45us floor at 23.3 TB/s.

typedef float v2f __attribute__((ext_vector_type(2)));
typedef float v8f __attribute__((ext_vector_type(8)));

#define NROWS 1536
#define DDIM  64
#define MINN  1e-15f
#define ATCLIP (1.0f - 1e-7f)
#define MAXNORM 0.996f   // (1 - PROJ_EPS)/sqrt(C)

__device__ __forceinline__ float artanh_clip(float y) {
  // y >= 0 on all call sites
  y = fminf(y, ATCLIP);
  return 0.5f * __logf((1.0f + y) / (1.0f - y));
}

// ---------------- Phase 0: per-row stats ----------------
// x2[i] = ||x_i||^2 ; tl[i] = logmap0(x_i)·w_l ; tr[i] = logmap0(x_i)·w_r
__global__ void __launch_bounds__(256) hyp_rowstats(
    const float* __restrict__ X, const float* __restrict__ attw,
    float* __restrict__ x2a, float* __restrict__ tla, float* __restrict__ tra) {
  int i = blockIdx.x * blockDim.x + threadIdx.x;
  if (i >= NROWS) return;
  float s2 = 0.f, dl = 0.f, dr = 0.f;
  const float* xr = X + (size_t)i * DDIM;
#pragma unroll 8
  for (int k = 0; k < DDIM; ++k) {
    float xv = xr[k];
    s2 += xv * xv;
    dl += xv * attw[k];
    dr += xv * attw[DDIM + k];
  }
  float xn = fmaxf(sqrtf(s2), MINN);
  float scale = artanh_clip(xn) / xn;   // logmap0 radial scale
  x2a[i] = s2;
  tla[i] = scale * dl;
  tra[i] = scale * dr;
}

// ---------------- Phase 1: fused Gram -> attention -> aggregation ----------------
// One workgroup = 16 output rows. 8 waves split the 96 j-tiles (12 each).
__global__ void __launch_bounds__(256) hyp_agg(
    const float* __restrict__ X, const float* __restrict__ adj,
    const float* __restrict__ x2a, const float* __restrict__ tla,
    const float* __restrict__ tra, const float* __restrict__ attb,
    float* __restrict__ out) {
  __shared__ float s_xj[8][16 * 65];   // per-wave X_j tile (16x64, pad 65)
  __shared__ float s_q[8][16 * 17];    // per-wave Q tile (16x16, pad 17)
  __shared__ float s_acc[16 * DDIM];   // cross-wave S accumulator
  __shared__ float s_prow[16];         // cross-wave rowP accumulator

  const int tid  = threadIdx.x;
  const int lane = tid & 31;
  const int wid  = tid >> 5;
  const int i0   = blockIdx.x * 16;
  const int nlo  = lane & 15;          // row/col index within 16
  const int hi   = lane >> 4;          // half-wave select
  const int koff = hi * 2;             // fp32 WMMA A/B frag: lanes16-31 hold K+2,K+3

  for (int t = tid; t < 16 * DDIM; t += 256) s_acc[t] = 0.f;
  if (tid < 16) s_prow[tid] = 0.f;

  // Preload Gram A fragments (rows i0..i0+15, K = d = 64 -> 16 k-steps of 4).
  // fp32 16x16x4 A layout: lane l holds A[l%16, kb + 2*(l/16) + {0,1}].
  v2f afr[16];
#pragma unroll
  for (int ks = 0; ks < 16; ++ks) {
    const float* p = X + (size_t)(i0 + nlo) * DDIM + ks * 4 + koff;
    afr[ks].x = p[0];
    afr[ks].y = p[1];
  }

  // Per-lane row-i constants: C/D layout puts row m = r + 8*hi in VGPR r.
  float x2row[8], tlrow[8];
#pragma unroll
  for (int r = 0; r < 8; ++r) {
    x2row[r] = x2a[i0 + r + 8 * hi];
    tlrow[r] = tla[i0 + r + 8 * hi];
  }
  const float bb = attb[0];

  v8f sacc0 = {}, sacc1 = {}, sacc2 = {}, sacc3 = {};
  float pacc[8];
#pragma unroll
  for (int r = 0; r < 8; ++r) pacc[r] = 0.f;

  float* xj = s_xj[wid];
  float* qb = s_q[wid];

  __syncthreads();

  const int NITER = (NROWS / 16) / 8;  // 12 j-tiles per wave (uniform -> barriers legal)
  for (int t = 0; t < NITER; ++t) {
    const int j0 = (wid + t * 8) * 16;

    // Stage X_j tile (16x64) into LDS, coalesced b64 per lane.
#pragma unroll
    for (int rr = 0; rr < 16; ++rr) {
      const float* src = X + (size_t)(j0 + rr) * DDIM + lane * 2;
      xj[rr * 65 + lane * 2 + 0] = src[0];
      xj[rr * 65 + lane * 2 + 1] = src[1];
    }

    // Prefetch next tile's adj block into cache while we compute this one.
    if (t + 1 < NITER) {
      const int j0n = (wid + (t + 1) * 8) * 16;
      __builtin_prefetch(adj + (size_t)(i0 + nlo) * NROWS + j0n, 0, 1);
    }

    __syncthreads();

    // ---- Gram tile: G(16x16) = X_i(16x64) * X_j^T(64x16), fp32 WMMA ----
    v8f G = {};
#pragma unroll
    for (int ks = 0; ks < 16; ++ks) {
      v2f b;   // B[k,n] = X[j0+n, k]: same per-lane gather pattern as A frag
      b.x = xj[nlo * 65 + ks * 4 + koff];
      b.y = xj[nlo * 65 + ks * 4 + koff + 1];
      G = __builtin_amdgcn_wmma_f32_16x16x4_f32(false, afr[ks], false, b,
                                                (short)0, G, false, false);
    }

    // ---- Element-wise: attention weight + logmap scalar coefficients ----
    const float y2j = x2a[j0 + nlo];
    const float trj = tra[j0 + nlo];
#pragma unroll
    for (int r = 0; r < 8; ++r) {
      const int m = r + 8 * hi;                    // C/D layout row
      const float g    = G[r];                     // x_i · x_j
      const float x2i  = x2row[r];
      const float cB   = 1.0f - x2i;               // B = 1 - C*x2
      const float A    = 1.0f - 2.0f * g + y2j;    // 1 + 2C*xy + C*y2 (xy = -g)
      const float den  = fmaxf(1.0f - 2.0f * g + x2i * y2j, MINN);
      const float sub2 = fmaxf(A * A * x2i - 2.0f * A * cB * g + cB * cB * y2j, 0.f);
      const float sn   = fmaxf(sqrtf(sub2) / den, MINN);
      const float coef = cB * artanh_clip(sn) / sn;          // 2/(lam)*artanh(sn)/sn
      const float logit = tlrow[r] + trj + bb;
      const float w = adj[(size_t)(i0 + m) * NROWS + j0 + nlo] /
                      (1.0f + __expf(-logit));               // sigmoid * adj
      const float tf = w * coef / den;
      pacc[r] += -tf * A;                                    // coeff on x_i
      qb[m * 17 + nlo] = tf * cB;                            // coeff on x_j
    }

    __syncthreads();   // relayout Q: C/D-layout -> A-fragment via LDS

    // ---- Aggregation: S(16x64) += Q(16x16) @ X_j(16x64), fp32 WMMA ----
#pragma unroll
    for (int ks = 0; ks < 4; ++ks) {
      v2f aq;
      aq.x = qb[nlo * 17 + ks * 4 + koff];
      aq.y = qb[nlo * 17 + ks * 4 + koff + 1];
      const float* r0 = xj + (ks * 4 + koff) * 65 + nlo;      // B[k,  n]
      const float* r1 = xj + (ks * 4 + koff + 1) * 65 + nlo;  // B[k+1,n]
      v2f b0, b1, b2, b3;
      b0.x = r0[0];  b0.y = r1[0];
      b1.x = r0[16]; b1.y = r1[16];
      b2.x = r0[32]; b2.y = r1[32];
      b3.x = r0[48]; b3.y = r1[48];
      sacc0 = __builtin_amdgcn_wmma_f32_16x16x4_f32(false, aq, false, b0, (short)0, sacc0, false, false);
      sacc1 = __builtin_amdgcn_wmma_f32_16x16x4_f32(false, aq, false, b1, (short)0, sacc1, false, false);
      sacc2 = __builtin_amdgcn_wmma_f32_16x16x4_f32(false, aq, false, b2, (short)0, sacc2, false, false);
      sacc3 = __builtin_amdgcn_wmma_f32_16x16x4_f32(false, aq, false, b3, (short)0, sacc3, false, false);
    }

    __syncthreads();   // keep xj/qb live until all reads done
  }

  // ---- Cross-wave reduction (LDS float atomics -> ds_add_f32) ----
#pragma unroll
  for (int r = 0; r < 8; ++r) {
    const int m = r + 8 * hi;
    atomicAdd(&s_prow[m], pacc[r]);
    atomicAdd(&s_acc[m * DDIM +  0 + nlo], sacc0[r]);
    atomicAdd(&s_acc[m * DDIM + 16 + nlo], sacc1[r]);
    atomicAdd(&s_acc[m * DDIM + 32 + nlo], sacc2[r]);
    atomicAdd(&s_acc[m * DDIM + 48 + nlo], sacc3[r]);
  }
  __syncthreads();

  // ---- Epilogue: u = S + rowP*x, then expmap(x,u) and proj (16 rows) ----
  if (tid < 16) {
    const int m = tid;
    const int i = i0 + m;
    const float x2i = x2a[i];
    const float pr = s_prow[m];
    float un2 = 0.f, xu = 0.f;
    for (int k = 0; k < DDIM; ++k) {
      float xv = X[(size_t)i * DDIM + k];
      float u = s_acc[m * DDIM + k] + pr * xv;
      s_acc[m * DDIM + k] = u;
      un2 += u * u;
      xu  += xv * u;
    }
    const float un  = fmaxf(sqrtf(un2), MINN);
    const float th  = tanhf(un / (1.0f - x2i));    // tanh(lam/2 * un)
    const float iun = 1.0f / un;
    const float xy  = th * xu * iun;               // x · second
    const float y2  = th * th;                     // ||second||^2
    const float dn2 = fmaxf(1.0f + 2.0f * xy + x2i * y2, MINN);
    const float cx  = (1.0f + 2.0f * xy + y2) / dn2;
    const float cu  = (1.0f - x2i) * th * iun / dn2;
    const float rs2 = cx * cx * x2i + 2.f * cx * cu * xu + cu * cu * un2;
    const float rn  = fmaxf(sqrtf(rs2), MINN);
    const float f   = (rn > MAXNORM) ? (MAXNORM / rn) : 1.0f;
    for (int k = 0; k < DDIM; ++k) {
      float xv = X[(size_t)i * DDIM + k];
      out[(size_t)i * DDIM + k] = f * (cx * xv + cu * s_acc[m * DDIM + k]);
    }
  }
}

extern "C" void kernel_launch(void* const* d_in, const int* in_sizes, int n_in,
                              void* d_out, int out_size, void* d_ws, size_t ws_size,
                              hipStream_t stream) {
  const float* X    = (const float*)d_in[0];   // (1536, 64)
  const float* adj  = (const float*)d_in[1];   // (1536, 1536)
  const float* attw = (const float*)d_in[2];   // (128, 1)
  const float* attb = (const float*)d_in[3];   // (1,)
  float* out = (float*)d_out;

  float* x2a = (float*)d_ws;                   // 3*1536 floats of scratch
  float* tla = x2a + NROWS;
  float* tra = tla + NROWS;

  hyp_rowstats<<<(NROWS + 255) / 256, 256, 0, stream>>>(X, attw, x2a, tla, tra);
  hyp_agg<<<NROWS / 16, 256, 0, stream>>>(X, adj, x2a, tla, tra, attb, out);
}